// QuantumFeatureExtractor_65481071398302
// MI455X (gfx1250) — compile-verified
//
#include <hip/hip_runtime.h>
#include <hip/hip_bf16.h>

typedef float v2f __attribute__((ext_vector_type(2)));
typedef float v8f __attribute__((ext_vector_type(8)));

// ---------------------------------------------------------------------------
// Setup kernel: build the fixed 16x16 circuit unitary U = Ur + i*Ui from the
// 6 layer params. One thread per basis column j. Qubit w <-> bit (3-w).
// Layout in ws: Ur[16][16] row-major (row = output index k), then Ui[16][16].
// ---------------------------------------------------------------------------
__global__ void qfe_build_unitary(const float* __restrict__ t, float* __restrict__ U) {
    int j = threadIdx.x;
    if (j >= 16) return;

    float cr[16], ci[16];
#pragma unroll
    for (int k = 0; k < 16; ++k) { cr[k] = (k == j) ? 1.f : 0.f; ci[k] = 0.f; }

    auto ap1q = [&](float g00r, float g00i, float g01r, float g01i,
                    float g10r, float g10i, float g11r, float g11i, int bit) {
        int mask = 1 << bit;
#pragma unroll
        for (int k = 0; k < 16; ++k) {
            if (k & mask) continue;
            int k2 = k | mask;
            float v0r = cr[k],  v0i = ci[k];
            float v1r = cr[k2], v1i = ci[k2];
            cr[k]  = g00r*v0r - g00i*v0i + g01r*v1r - g01i*v1i;
            ci[k]  = g00r*v0i + g00i*v0r + g01r*v1i + g01i*v1r;
            cr[k2] = g10r*v0r - g10i*v0i + g11r*v1r - g11i*v1i;
            ci[k2] = g10r*v0i + g10i*v0r + g11r*v1i + g11i*v1r;
        }
    };
    auto cnot = [&](int cbit, int tbit) {
        int cm = 1 << cbit, tm = 1 << tbit;
#pragma unroll
        for (int k = 0; k < 16; ++k) {
            if ((k & cm) && !(k & tm)) {
                int k2 = k | tm;
                float tr = cr[k], ti = ci[k];
                cr[k] = cr[k2]; ci[k] = ci[k2];
                cr[k2] = tr;    ci[k2] = ti;
            }
        }
    };

    float c, s;
    // RX(t0) on qubit 0 (bit 3): [[c, -i s], [-i s, c]]
    c = cosf(0.5f * t[0]); s = sinf(0.5f * t[0]);
    ap1q(c, 0.f, 0.f, -s, 0.f, -s, c, 0.f, 3);
    // RY(t1) on qubit 1 (bit 2): [[c, -s], [s, c]]
    c = cosf(0.5f * t[1]); s = sinf(0.5f * t[1]);
    ap1q(c, 0.f, -s, 0.f, s, 0.f, c, 0.f, 2);
    // RZ(t2) on qubit 2 (bit 1): diag(e^{-it/2}, e^{+it/2})
    c = cosf(0.5f * t[2]); s = sinf(0.5f * t[2]);
    ap1q(c, -s, 0.f, 0.f, 0.f, 0.f, c, s, 1);
    // CNOT control q0 (bit 3), target q2 (bit 1)
    cnot(3, 1);
    // RX(t3) on qubit 3 (bit 0)
    c = cosf(0.5f * t[3]); s = sinf(0.5f * t[3]);
    ap1q(c, 0.f, 0.f, -s, 0.f, -s, c, 0.f, 0);
    // RY(t4) on qubit 0 (bit 3)
    c = cosf(0.5f * t[4]); s = sinf(0.5f * t[4]);
    ap1q(c, 0.f, -s, 0.f, s, 0.f, c, 0.f, 3);
    // CNOT control q1 (bit 2), target q3 (bit 0)
    cnot(2, 0);
    // RZ(t5) on qubit 2 (bit 1)
    c = cosf(0.5f * t[5]); s = sinf(0.5f * t[5]);
    ap1q(c, -s, 0.f, 0.f, 0.f, 0.f, c, s, 1);

    float* Ur = U;
    float* Ui = U + 256;
#pragma unroll
    for (int k = 0; k < 16; ++k) {
        Ur[k * 16 + j] = cr[k];
        Ui[k * 16 + j] = ci[k];
    }
}

// ---------------------------------------------------------------------------
// Main kernel: one wave processes groups of 16 consecutive patches.
//   1) Y = U * S  (complex): 8x v_wmma_f32_16x16x4_f32 (4 K-slices x re/im)
//   2) P = |Y|^2  (elementwise VALU)
//   3) meas = Z * P : 4 more WMMAs with the constant +-1 sign matrix Z
//      (padded to 16 rows), so D[M=w][N=patch] and low-half lanes hold the
//      four measurement values in VGPRs 0..3 -> one coalesced b128 store.
//
// 32-bit A-matrix 16x4 layout: lane l holds row M=l&15; VGPR v holds
// K = 4*kk + 2h + v with h = l>>4. B 4x16 mirrors this (N = l&15).
// C/D: lane l holds D[M = r + 8h][N = l&15] in VGPR r.
// ---------------------------------------------------------------------------
__global__ __launch_bounds__(256)
void qfe_main(const float* __restrict__ x, const float* __restrict__ U,
              float* __restrict__ out, int groups_per_wave, int total_groups) {
    const int lane = threadIdx.x & 31;
    const int wave = blockIdx.x * (blockDim.x >> 5) + (threadIdx.x >> 5);
    const int h    = lane >> 4;      // lane half
    const int m16  = lane & 15;      // A row / patch column / output patch

    const float* Ur = U;
    const float* Ui = U + 256;

    // Preload per-lane A fragments of Ur and Ui (one 8-byte load each slice).
    v2f aur[4], aui[4];
#pragma unroll
    for (int kk = 0; kk < 4; ++kk) {
        int off = m16 * 16 + 4 * kk + 2 * h;
        aur[kk] = *(const v2f*)(Ur + off);
        aui[kk] = *(const v2f*)(Ui + off);
    }

    // Constant A fragments of the padded sign matrix Z (16x16, rows 0..3 are
    // the four qubit Z-parity rows, rows 4..15 zero). Row w = m16, col m.
    v2f azf[4];
#pragma unroll
    for (int kk = 0; kk < 4; ++kk) {
#pragma unroll
        for (int v = 0; v < 2; ++v) {
            int m = 4 * kk + 2 * h + v;
            float val = 0.f;
            if (m16 < 4) val = ((m >> (3 - m16)) & 1) ? -1.f : 1.f;
            azf[kk][v] = val;
        }
    }

    const int g0 = wave * groups_per_wave;
    for (int gi = 0; gi < groups_per_wave; ++gi) {
        int g = g0 + gi;
        if (g >= total_groups) break;           // wave-uniform

        int n  = g * 16 + m16;                  // patch index for this lane
        int b  = n >> 12;                       // image
        int ph = (n >> 6) & 63;                 // patch row
        int pw = n & 63;                        // patch col

        // angles: x[b, 2ph + r, 2pw + c], image is 128x128
        const float* px = x + ((size_t)b << 14) + ((size_t)ph << 8) + (pw << 1);
        v2f r0 = *(const v2f*)px;               // ang0, ang1
        v2f r1 = *(const v2f*)(px + 128);       // ang2, ang3

        float s0, c0, s1, c1, s2, c2, s3, c3;
        __sincosf(0.5f * r0.x, &s0, &c0);
        __sincosf(0.5f * r0.y, &s1, &c1);
        __sincosf(0.5f * r1.x, &s2, &c2);
        __sincosf(0.5f * r1.y, &s3, &c3);

        // State index j = 4*kk + 2h + v: bit3=kk>>1 (q0), bit2=kk&1 (q1),
        // bit1=h (q2), bit0=v (q3). s[j] = A0[b3]*A1[b2]*A2[b1]*A3[b0].
        float t2 = h ? s2 : c2;

        v8f yr = {};
        v8f yi = {};
#pragma unroll
        for (int kk = 0; kk < 4; ++kk) {
            float sel0 = (kk & 2) ? s0 : c0;
            float sel1 = (kk & 1) ? s1 : c1;
            float mm   = sel0 * sel1 * t2;
            v2f bfrag  = { mm * c3, mm * s3 };
            yr = __builtin_amdgcn_wmma_f32_16x16x4_f32(
                     false, aur[kk], false, bfrag, (short)0, yr, false, false);
            yi = __builtin_amdgcn_wmma_f32_16x16x4_f32(
                     false, aui[kk], false, bfrag, (short)0, yi, false, false);
        }

        // p[r] = |Y[m = r + 8h][patch]|^2
        float p[8];
#pragma unroll
        for (int r = 0; r < 8; ++r) p[r] = yr[r] * yr[r] + yi[r] * yi[r];

        // Build B fragments of P for the measurement WMMA. Lane needs p[m]
        // for m with bit1(m)=h; it holds m with bit3(m)=h. Exchange the four
        // non-local values with the partner lane (lane ^ 16).
        //   h=0 sends p[2],p[3],p[6],p[7]; h=1 sends p[0],p[1],p[4],p[5].
        float q0 = h ? p[0] : p[2];
        float q1 = h ? p[1] : p[3];
        float q2 = h ? p[4] : p[6];
        float q3 = h ? p[5] : p[7];
        float e0 = __shfl_xor(q0, 16, 32);
        float e1 = __shfl_xor(q1, 16, 32);
        float e2 = __shfl_xor(q2, 16, 32);
        float e3 = __shfl_xor(q3, 16, 32);

        v2f bp[4];
        bp[0] = h ? (v2f){e0, e1} : (v2f){p[0], p[1]};   // m = {2h, 2h+1}
        bp[1] = h ? (v2f){e2, e3} : (v2f){p[4], p[5]};   // m = {4+2h, 5+2h}
        bp[2] = h ? (v2f){p[2], p[3]} : (v2f){e0, e1};   // m = {8+2h, 9+2h}
        bp[3] = h ? (v2f){p[6], p[7]} : (v2f){e2, e3};   // m = {12+2h, 13+2h}

        v8f d = {};
#pragma unroll
        for (int kk = 0; kk < 4; ++kk) {
            d = __builtin_amdgcn_wmma_f32_16x16x4_f32(
                    false, azf[kk], false, bp[kk], (short)0, d, false, false);
        }

        // Low-half lanes hold meas[w=0..3][patch m16] in d[0..3].
        if (h == 0) {
            float4 meas;
            meas.x = d[0];
            meas.y = d[1];
            meas.z = d[2];
            meas.w = d[3];
            *(float4*)(out + (size_t)n * 4) = meas;
        }
    }
}

extern "C" void kernel_launch(void* const* d_in, const int* in_sizes, int n_in,
                              void* d_out, int out_size, void* d_ws, size_t ws_size,
                              hipStream_t stream) {
    const float* x = (const float*)d_in[0];       // (128,1,128,128) f32
    const float* t = (const float*)d_in[1];       // (6,) f32
    float* out = (float*)d_out;                   // (128, 16384) f32
    float* U   = (float*)d_ws;                    // Ur[256] + Ui[256]

    hipLaunchKernelGGL(qfe_build_unitary, dim3(1), dim3(32), 0, stream, t, U);

    int npatch       = in_sizes[0] / 4;           // 524288
    int total_groups = npatch / 16;               // 32768
    const int gpw    = 8;                         // groups per wave
    int waves  = (total_groups + gpw - 1) / gpw;  // 4096
    int blocks = (waves + 7) / 8;                 // 8 waves per 256-thread block

    hipLaunchKernelGGL(qfe_main, dim3(blocks), dim3(256), 0, stream,
                       x, U, out, gpw, total_groups);
}